// patientTripletLoss_52536039964791
// MI455X (gfx1250) — compile-verified
//
#include <hip/hip_runtime.h>
#include <stdint.h>

#define B_N   8192
#define D_F   512
#define TILE  2048
#define EPSF  1e-7f

typedef __attribute__((ext_vector_type(4))) unsigned int uint32x4;
typedef __attribute__((ext_vector_type(8))) int          int32x8;
typedef __attribute__((ext_vector_type(4))) int          int32x4;
typedef __attribute__((ext_vector_type(2))) float        v2f;
typedef __attribute__((ext_vector_type(4))) float        v4f;
typedef __attribute__((ext_vector_type(8))) float        v8f;

// ---------------------------------------------------------------------------
// TDM: DMA one contiguous TILE-float run from global into LDS (CDNA5 path).
// Descriptor per CDNA5 ISA. 6-arg builtin on this
// toolchain: (uint32x4 g0, int32x8 g1, int32x4 g2, int32x4 g3, int32x8, i32).
// ---------------------------------------------------------------------------
__device__ __forceinline__ void tdm_load_tile(const float* gsrc, void* ldsDst) {
  uint64_t ga = (uint64_t)(uintptr_t)gsrc;
  uint32_t la = (uint32_t)(uintptr_t)ldsDst;   // low 32 bits of flat shared = LDS offset
  uint32x4 g0 = {
    1u,                                                  // count=1, user mode
    la,                                                  // lds_addr[31:0]
    (uint32_t)(ga & 0xFFFFFFFFu),                        // global_addr[31:0]
    (uint32_t)((ga >> 32) & 0x01FFFFFFu) | (2u << 30)    // global_addr[56:32] | type=2
  };
  int32x8 g1 = {
    0x00020000,                                          // data_size=2 (4 bytes)
    (int)(((unsigned)TILE & 0xFFFFu) << 16),             // tensor_dim0[15:0]
    (int)(((unsigned)TILE >> 16) | (1u << 16)),          // tensor_dim0[31:16] | tensor_dim1=1
    (int)((unsigned)TILE << 16),                         // tile_dim0 = TILE
    1,                                                   // tile_dim1 = 1, tile_dim2 = 0
    (int)TILE,                                           // tensor_dim0_stride[31:0]
    0, 0
  };
  int32x4 gz4 = {0, 0, 0, 0};
  int32x8 gz8 = {0, 0, 0, 0, 0, 0, 0, 0};
  __builtin_amdgcn_tensor_load_to_lds(g0, g1, gz4, gz4, gz8, 0);
}

// ---------------------------------------------------------------------------
// Kernel 0: pack labels into a 8192-bit mask (1 KB).
// ---------------------------------------------------------------------------
__global__ void pack_labels_kernel(const int* __restrict__ labels,
                                   uint32_t* __restrict__ bitmask) {
  int w = threadIdx.x;  // 256 words
  uint32_t m = 0;
#pragma unroll
  for (int b = 0; b < 32; ++b)
    m |= (uint32_t)(labels[w * 32 + b] & 1) << b;
  bitmask[w] = m;
}

// ---------------------------------------------------------------------------
// Kernel 1: per-row masked argmax over noise[0] (same-label) and noise[1]
// (opposite-label). One workgroup per row; noise streamed via TDM
// double-buffered 8 KB tiles. ~512 MB total => HBM-bound (~22 us).
// Inner loop: 2 x ds_load_b128 per thread per tile (lane-interleaved float4s).
// ---------------------------------------------------------------------------
__global__ void argmax_kernel(const float* __restrict__ noise,
                              const uint32_t* __restrict__ bitmask,
                              const int* __restrict__ labels,
                              int* __restrict__ posIdx,
                              int* __restrict__ negIdx) {
  __shared__ __align__(16) float buf[2][TILE];   // 16 KB double buffer
  __shared__ uint32_t lmask[B_N / 32];           // 1 KB label bits
  __shared__ float    redV[256];
  __shared__ int      redI[256];

  const int row = blockIdx.x;
  const int tid = threadIdx.x;

  lmask[tid] = bitmask[tid];
  const int myLab = labels[row] & 1;

  const float* rowPos = noise + (size_t)row * B_N;
  const float* rowNeg = noise + (size_t)B_N * B_N + (size_t)row * B_N;

  // Prime the pipeline: wave 0 issues the first two tiles (TDM ignores EXEC,
  // so gate at wave granularity via control flow).
  if (tid < 32) {
    tdm_load_tile(rowPos, &buf[0][0]);
    tdm_load_tile(rowPos + TILE, &buf[1][0]);
  }

  float bestV[2] = {-2.0f, -2.0f};
  int   bestI[2] = {0, 0};

  // Tiles 0..3 = noise[0] row, tiles 4..7 = noise[1] row.
  for (int t = 0; t < 8; ++t) {
    const int plane = t >> 2;
    const int base  = (t & 3) * TILE;

    if (tid < 32) {
      if (t < 7) __builtin_amdgcn_s_wait_tensorcnt(1);  // tile t complete
      else       __builtin_amdgcn_s_wait_tensorcnt(0);
    }
    __syncthreads();  // publish TDM-written LDS to all waves

    const v4f* src4 = (const v4f*)&buf[t & 1][0];
#pragma unroll
    for (int k = 0; k < 2; ++k) {
      const int   f  = k * 256 + tid;          // float4 index within tile
      const v4f   v  = src4[f];
      const int   j0 = base + f * 4;           // first column of this quad
      const uint32_t w = lmask[j0 >> 5] >> (j0 & 31);  // 4 bits, no straddle
#pragma unroll
      for (int e = 0; e < 4; ++e) {
        const bool  match = ((int)((w >> e) & 1u) ^ myLab) == plane;
        const float mv = match ? v[e] : -1.0f;
        if (mv > bestV[plane]) { bestV[plane] = mv; bestI[plane] = j0 + e; }
      }
    }
    __syncthreads();  // all waves done reading buf[t&1] before reissue

    const int nt = t + 2;
    if (tid < 32 && nt < 8) {
      const float* rp = (nt >= 4) ? rowNeg : rowPos;
      tdm_load_tile(rp + (size_t)(nt & 3) * TILE, &buf[nt & 1][0]);
    }
  }

  // Block-wide (val, lowest-index) argmax reduction, matching jnp.argmax ties.
  for (int plane = 0; plane < 2; ++plane) {
    redV[tid] = bestV[plane];
    redI[tid] = bestI[plane];
    __syncthreads();
    for (int s = 128; s > 0; s >>= 1) {
      if (tid < s) {
        const float v2 = redV[tid + s];
        const int   i2 = redI[tid + s];
        if (v2 > redV[tid] || (v2 == redV[tid] && i2 < redI[tid])) {
          redV[tid] = v2; redI[tid] = i2;
        }
      }
      __syncthreads();
    }
    if (tid == 0) {
      if (plane == 0) posIdx[row] = redI[0];
      else            negIdx[row] = (redV[0] <= -1.0f) ? row : redI[0];  // no-neg fallback
    }
    __syncthreads();
  }
}

// ---------------------------------------------------------------------------
// Kernel 2: per-row triplet loss via WMMA. One wave handles 16 rows: the
// diagonals of diff*diff^T (accumulated by V_WMMA_F32_16X16X4_F32 over k)
// are the squared L2 norms of (a - p + eps) / (a - n + eps).
// f32 A(16x4)/B(4x16) fragments coincide per lane and equal the lane's own
// float2 at k0 + 2h, so fragments come straight from the loads (no selects).
// ---------------------------------------------------------------------------
__global__ void triplet_wmma_kernel(const float* __restrict__ feats,
                                    const int* __restrict__ posIdx,
                                    const int* __restrict__ negIdx,
                                    float* __restrict__ losses) {
  const int wave = threadIdx.x >> 5;                   // 8 waves / block
  const int lane = threadIdx.x & 31;
  const int tileRow = (blockIdx.x * 8 + wave) * 16;    // 16 rows per wave
  const int r = tileRow + (lane & 15);
  const int h = lane >> 4;                             // lane half owns K = 2h, 2h+1

  const float* aRow = feats + (size_t)r * D_F + 2 * h;
  const float* pRow = feats + (size_t)posIdx[r] * D_F + 2 * h;
  const float* nRow = feats + (size_t)negIdx[r] * D_F + 2 * h;

  v8f accP = {};
  v8f accN = {};

#pragma unroll 4
  for (int k0 = 0; k0 < D_F; k0 += 4) {
    const v2f av = *(const v2f*)(aRow + k0);
    const v2f pv = *(const v2f*)(pRow + k0);
    const v2f nv = *(const v2f*)(nRow + k0);
    const v2f fP = av - pv + EPSF;
    const v2f fN = av - nv + EPSF;

    accP = __builtin_amdgcn_wmma_f32_16x16x4_f32(false, fP, false, fP,
                                                 (short)0, accP, false, false);
    accN = __builtin_amdgcn_wmma_f32_16x16x4_f32(false, fN, false, fN,
                                                 (short)0, accN, false, false);
  }

  // Extract diagonal: D[m][m] sits at (vgpr m, lane m) for m<8 and
  // (vgpr m-8, lane m+16) for m>=8.
  float dP = 0.0f, dN = 0.0f;
#pragma unroll
  for (int v = 0; v < 8; ++v) {
    if (lane == v)      { dP = accP[v]; dN = accN[v]; }
    if (lane == v + 24) { dP = accP[v]; dN = accN[v]; }
  }
  const float dP2 = __shfl(dP, lane + 16, 32);  // lanes 8..15 fetch 24..31
  const float dN2 = __shfl(dN, lane + 16, 32);
  if (lane >= 8 && lane < 16) { dP = dP2; dN = dN2; }

  if (lane < 16) {
    const float loss = sqrtf(dP) - sqrtf(dN) + 1.0f;  // margin = 1
    losses[r] = loss > 0.0f ? loss : 0.0f;
  }
}

// ---------------------------------------------------------------------------
// Kernel 3: deterministic fixed-order mean reduction -> d_out[0].
// ---------------------------------------------------------------------------
__global__ void reduce_mean_kernel(const float* __restrict__ losses,
                                   float* __restrict__ out) {
  __shared__ float s[256];
  const int tid = threadIdx.x;
  float acc = 0.0f;
  for (int i = tid; i < B_N; i += 256) acc += losses[i];
  s[tid] = acc;
  __syncthreads();
  for (int st = 128; st > 0; st >>= 1) {
    if (tid < st) s[tid] += s[tid + st];
    __syncthreads();
  }
  if (tid == 0) out[0] = s[0] * (1.0f / (float)B_N);
}

// ---------------------------------------------------------------------------
extern "C" void kernel_launch(void* const* d_in, const int* in_sizes, int n_in,
                              void* d_out, int out_size, void* d_ws, size_t ws_size,
                              hipStream_t stream) {
  (void)in_sizes; (void)n_in; (void)out_size; (void)ws_size;

  const float* feats  = (const float*)d_in[0];   // [8192, 512] f32
  const int*   labels = (const int*)d_in[1];     // [8192] int
  const float* noise  = (const float*)d_in[2];   // [2, 8192, 8192] f32
  float*       out    = (float*)d_out;           // scalar f32

  uint32_t* bitmask = (uint32_t*)d_ws;                 // 256 words
  int*      posIdx  = (int*)(bitmask + 256);           // 8192 ints
  int*      negIdx  = posIdx + B_N;                    // 8192 ints
  float*    losses  = (float*)(negIdx + B_N);          // 8192 floats

  pack_labels_kernel<<<1, 256, 0, stream>>>(labels, bitmask);
  argmax_kernel<<<B_N, 256, 0, stream>>>(noise, bitmask, labels, posIdx, negIdx);
  triplet_wmma_kernel<<<B_N / 16 / 8, 256, 0, stream>>>(feats, posIdx, negIdx, losses);
  reduce_mean_kernel<<<1, 256, 0, stream>>>(losses, out);
}